// PaiNN_82308753261028
// MI455X (gfx1250) — compile-verified
//
#include <hip/hip_runtime.h>
#include <hip/hip_bf16.h>

typedef __attribute__((ext_vector_type(2))) float v2f;
typedef __attribute__((ext_vector_type(8))) float v8f;

#define NF   128
#define NPHI 384
#define NRBF 20

__device__ __forceinline__ v8f wmma4(v2f a, v2f b, v8f c) {
  // D = A(16x4,f32) * B(4x16,f32) + C(16x16,f32)
  return __builtin_amdgcn_wmma_f32_16x16x4_f32(false, a, false, b, (short)0, c,
                                               false, false);
}

__device__ __forceinline__ float fast_silu(float h) {
  // h * sigmoid(h); hardware v_rcp_f32 instead of IEEE division sequence
  return h * __builtin_amdgcn_rcpf(1.0f + __expf(-h));
}

// -------- init: out_scalar = scalar_features, out_vector = vector_features ----
__global__ void painn_init_kernel(const float4* __restrict__ s,
                                  const float4* __restrict__ v,
                                  float4* __restrict__ out, int ns4, int nv4) {
  const int stride = gridDim.x * blockDim.x;
  for (int k = blockIdx.x * blockDim.x + threadIdx.x; k < ns4; k += stride)
    out[k] = s[k];
  for (int k = blockIdx.x * blockDim.x + threadIdx.x; k < nv4; k += stride)
    out[ns4 + k] = v[k];
}

// -------- fused node MLP: phi = Linear2(silu(Linear1(x))) --------------------
// one block (8 waves) per 16-node tile
__global__ __launch_bounds__(256) void painn_phi_kernel(
    const float* __restrict__ x, const float* __restrict__ W1,
    const float* __restrict__ b1, const float* __restrict__ W2,
    const float* __restrict__ b2, float* __restrict__ phi, int n_nodes) {
  __shared__ float hs[16][NF + 4];   // padded: stride 132 -> conflict-free

  const int tid   = threadIdx.x;
  const int wave  = tid >> 5;
  const int lane  = tid & 31;
  const int mrow  = lane & 15;
  const int khalf = lane >> 4;       // 0: K pair {0,1}, 1: K pair {2,3}
  const int mbase = khalf * 8;       // C/D row split at lane 16
  const int row0  = blockIdx.x * 16;
  const int rowA  = min(row0 + mrow, n_nodes - 1);

  // ---- layer 1: wave w computes h columns [16w, 16w+16) ----
  v8f acc = {};
  const int ncol = wave * 16 + mrow;
  #pragma unroll 4
  for (int k0 = 0; k0 < NF; k0 += 4) {
    const int ka = k0 + khalf * 2;
    v2f a, b;
    a.x = x[rowA * NF + ka];
    a.y = x[rowA * NF + ka + 1];
    b.x = W1[ncol * NF + ka];        // B[k][n] = W1[n][k]
    b.y = W1[ncol * NF + ka + 1];
    acc = wmma4(a, b, acc);
  }
  {
    const float bias = b1[ncol];
    #pragma unroll
    for (int v = 0; v < 8; ++v) {
      hs[mbase + v][ncol] = fast_silu(acc[v] + bias);
    }
  }
  __syncthreads();

  // ---- layer 2: 24 output tiles, 3 per wave ----
  for (int t = wave; t < 24; t += 8) {
    v8f acc2 = {};
    const int n = t * 16 + mrow;
    #pragma unroll 4
    for (int k0 = 0; k0 < NF; k0 += 4) {
      const int ka = k0 + khalf * 2;
      v2f a, b;
      a.x = hs[mrow][ka];
      a.y = hs[mrow][ka + 1];
      b.x = W2[n * NF + ka];
      b.y = W2[n * NF + ka + 1];
      acc2 = wmma4(a, b, acc2);
    }
    const float bias = b2[n];
    #pragma unroll
    for (int v = 0; v < 8; ++v) {
      const int m = row0 + mbase + v;
      if (m < n_nodes) phi[m * NPHI + n] = acc2[v] + bias;
    }
  }
}

// -------- edge kernel: RBF WMMA -> gate -> gather -> atomic scatter ----------
// one block (8 waves) per 16-edge tile
__global__ __launch_bounds__(256) void painn_edge_kernel(
    const int* __restrict__ idx_i, const int* __restrict__ idx_j,
    const float* __restrict__ rel_dir, const float* __restrict__ cut,
    const float* __restrict__ rbf, const float* __restrict__ Wr,
    const float* __restrict__ br, const float* __restrict__ phi,
    const float* __restrict__ vecf, float* __restrict__ out_s,
    float* __restrict__ out_v, int n_edges) {
  __shared__ float Wm[16][NPHI + 8];
  __shared__ float rbf_s[16][NRBF];
  __shared__ float dir_s[16][3];
  __shared__ float cut_s[16];
  __shared__ int   ii_s[16], jj_s[16];

  const int tid = threadIdx.x;
  const int e0  = blockIdx.x * 16;

  if (tid < 16) {
    const int e = min(e0 + tid, n_edges - 1);
    ii_s[tid] = idx_i[e];
    jj_s[tid] = idx_j[e];
    cut_s[tid] = cut[e];
    dir_s[tid][0] = rel_dir[e * 3 + 0];
    dir_s[tid][1] = rel_dir[e * 3 + 1];
    dir_s[tid][2] = rel_dir[e * 3 + 2];
  }
  for (int q = tid; q < 16 * NRBF; q += 256) {
    const int el = q / NRBF, k = q % NRBF;
    const int e = min(e0 + el, n_edges - 1);
    rbf_s[el][k] = rbf[e * NRBF + k];
  }
  __syncthreads();

  const int wave  = tid >> 5;
  const int lane  = tid & 31;
  const int mrow  = lane & 15;
  const int khalf = lane >> 4;
  const int mbase = khalf * 8;

  // W tile = (rbf[16x20] @ Wr.T + br) * cut  -> LDS
  for (int t = wave; t < 24; t += 8) {
    v8f acc = {};
    const int n = t * 16 + mrow;
    #pragma unroll
    for (int k0 = 0; k0 < NRBF; k0 += 4) {   // 20 = 5 * K4, exact
      const int ka = k0 + khalf * 2;
      v2f a, b;
      a.x = rbf_s[mrow][ka];
      a.y = rbf_s[mrow][ka + 1];
      b.x = Wr[n * NRBF + ka];               // B[k][n] = Wr[n][k]
      b.y = Wr[n * NRBF + ka + 1];
      acc = wmma4(a, b, acc);
    }
    const float bias = br[n];
    #pragma unroll
    for (int v = 0; v < 8; ++v) {
      const int m = mbase + v;
      Wm[m][n] = (acc[v] + bias) * cut_s[m];
    }
  }
  __syncthreads();

  // gather + gate + scatter: 2 edges x 128 features per pass
  const int f    = tid & 127;
  const int half = tid >> 7;
  for (int sub = 0; sub < 8; ++sub) {
    const int el = sub * 2 + half;
    if (e0 + el >= n_edges) continue;
    const int i = ii_s[el], j = jj_s[el];
    const float pvv = phi[j * NPHI + f]       * Wm[el][f];
    const float pss = phi[j * NPHI + 128 + f] * Wm[el][128 + f];
    const float pvs = phi[j * NPHI + 256 + f] * Wm[el][256 + f];
    atomicAdd(&out_s[i * NF + f], pss);
    const float dx = dir_s[el][0], dy = dir_s[el][1], dz = dir_s[el][2];
    const float* vj = vecf + ((size_t)j * NF + f) * 3;
    float* ov = out_v + ((size_t)i * NF + f) * 3;
    atomicAdd(&ov[0], vj[0] * pvv + pvs * dx);
    atomicAdd(&ov[1], vj[1] * pvv + pvs * dy);
    atomicAdd(&ov[2], vj[2] * pvv + pvs * dz);
  }
}

extern "C" void kernel_launch(void* const* d_in, const int* in_sizes, int n_in,
                              void* d_out, int out_size, void* d_ws,
                              size_t ws_size, hipStream_t stream) {
  const int*   idx_i   = (const int*)d_in[0];
  const int*   idx_j   = (const int*)d_in[1];
  const float* rel_dir = (const float*)d_in[2];
  const float* cutd    = (const float*)d_in[3];
  const float* rbf     = (const float*)d_in[4];
  const float* x       = (const float*)d_in[5];
  const float* vecf    = (const float*)d_in[6];
  const float* W1      = (const float*)d_in[7];
  const float* b1      = (const float*)d_in[8];
  const float* W2      = (const float*)d_in[9];
  const float* b2      = (const float*)d_in[10];
  const float* Wr      = (const float*)d_in[11];
  const float* br      = (const float*)d_in[12];

  const int n_edges = in_sizes[0];
  const int n_nodes = in_sizes[5] / NF;

  float* phi   = (float*)d_ws;                       // [N, 384]
  float* out_s = (float*)d_out;                      // [N, 128]
  float* out_v = out_s + (size_t)n_nodes * NF;       // [N, 128, 3]

  {
    const int ns4 = n_nodes * NF / 4;
    const int nv4 = n_nodes * NF * 3 / 4;
    painn_init_kernel<<<2048, 256, 0, stream>>>((const float4*)x,
                                                (const float4*)vecf,
                                                (float4*)d_out, ns4, nv4);
  }
  {
    const int nblocks = (n_nodes + 15) / 16;
    painn_phi_kernel<<<nblocks, 256, 0, stream>>>(x, W1, b1, W2, b2, phi,
                                                  n_nodes);
  }
  {
    const int nblocks = (n_edges + 15) / 16;
    painn_edge_kernel<<<nblocks, 256, 0, stream>>>(idx_i, idx_j, rel_dir, cutd,
                                                   rbf, Wr, br, phi, vecf,
                                                   out_s, out_v, n_edges);
  }
}